// DeepSAGE_31662498906634
// MI455X (gfx1250) — compile-verified
//
#include <hip/hip_runtime.h>

typedef float v2f __attribute__((ext_vector_type(2)));
typedef float v8f __attribute__((ext_vector_type(8)));

#define NNODE 50000
#define NEDGE 800000
#define DIN_  256
#define HDIM  384
#define NLAY  6
#define NCLS_ 2
#define EPS_  1e-5f
#define TK    16

// ------------------------------------------------ CDNA5 async global->LDS
__device__ __forceinline__ void async_ld_lds_b32(unsigned ldsOff, const float* gp) {
  // GLOBAL_LOAD_ASYNC_TO_LDS_B32: vdst = VGPR holding LDS byte address,
  // vaddr = 64-bit global address, no SADDR.  Tracked by ASYNCcnt.
  asm volatile("global_load_async_to_lds_b32 %0, %1, off"
               :: "v"(ldsOff), "v"(gp) : "memory");
}
__device__ __forceinline__ void wait_async0() {
  asm volatile("s_wait_asynccnt 0" ::: "memory");
}
__device__ __forceinline__ unsigned lds_off(const void* p) {
  // LDS generic address keeps the LDS byte offset in addr[31:0] (ISA 10.2).
  return (unsigned)(size_t)p;
}

// ---------------------------------------------------------------- utilities
__global__ void k_fill(float* __restrict__ p, float v, long n) {
  long i = (long)blockIdx.x * blockDim.x + threadIdx.x;
  long stride = (long)gridDim.x * blockDim.x;
  for (; i < n; i += stride) p[i] = v;
}

__global__ void k_degree(const int* __restrict__ col, float* __restrict__ cnt, int E) {
  int i = blockIdx.x * blockDim.x + threadIdx.x;
  if (i < E) unsafeAtomicAdd(&cnt[col[i]], 1.0f);
}

__global__ void k_invcnt(float* __restrict__ cnt, int N) {
  int i = blockIdx.x * blockDim.x + threadIdx.x;
  if (i < N) cnt[i] = 1.0f / fmaxf(cnt[i], 1.0f);
}

// scatter-add h[row[e], :] into agg[col[e], :]  (mean-scale folded into GEMM A-load)
__global__ void k_scatter(const float* __restrict__ h, const int* __restrict__ row,
                          const int* __restrict__ col, float* __restrict__ agg, int E) {
  long idx = (long)blockIdx.x * blockDim.x + threadIdx.x;
  long total = (long)E * (HDIM / 4);
  if (idx >= total) return;
  int e = (int)(idx / (HDIM / 4));
  int f = (int)(idx % (HDIM / 4)) * 4;
  const float4 v = *(const float4*)(h + (long)row[e] * HDIM + f);
  float* dst = agg + (long)col[e] * HDIM + f;
  unsafeAtomicAdd(dst + 0, v.x);
  unsafeAtomicAdd(dst + 1, v.y);
  unsafeAtomicAdd(dst + 2, v.z);
  unsafeAtomicAdd(dst + 3, v.w);
}

// ------------------------------------------------------------- batch norm
__global__ void k_bn_stats(const float* __restrict__ h, float* __restrict__ stats, int N) {
  int j = threadIdx.x;              // blockDim.x == HDIM (coalesced across columns)
  int r0 = blockIdx.x * 512;
  int r1 = min(r0 + 512, N);
  float s = 0.f, sq = 0.f;
  for (int r = r0; r < r1; ++r) {
    float v = h[(long)r * HDIM + j];
    s += v; sq += v * v;
  }
  unsafeAtomicAdd(&stats[j], s);
  unsafeAtomicAdd(&stats[HDIM + j], sq);
}

__global__ void k_bn_apply_relu(float* __restrict__ h, const float* __restrict__ stats,
                                const float* __restrict__ g, const float* __restrict__ b,
                                int N) {
  long n = (long)N * HDIM;
  long i = (long)blockIdx.x * blockDim.x + threadIdx.x;
  long stride = (long)gridDim.x * blockDim.x;
  const float invN = 1.0f / (float)N;
  for (; i < n; i += stride) {
    int j = (int)(i % HDIM);
    float mu  = stats[j] * invN;
    float var = stats[HDIM + j] * invN - mu * mu;
    float v = (h[i] - mu) * rsqrtf(var + EPS_) * g[j] + b[j];
    h[i] = fmaxf(v, 0.f);
  }
}

// -------------------------------------------------- generic WMMA f32 GEMM
// D[M,Ncol] = A1[M,K]@B1[K,Ncol] (+ A2@B2) + bias (+ Cin), optional ReLU.
// LDS layouts are chosen so every WMMA fragment is ONE aligned ds_load_b64:
//   A: row pitch 18 floats (even)  -> (As[m][k], As[m][k+1]) is an aligned pair
//   B: K-interleaved Bs[k/2][n][2] -> (B[k][n], B[k+1][n]) is an aligned pair
// Block: 128 threads = 4 waves; block tile 64x64, wave strip 16x64.
template <bool DUAL, bool RSCALE, bool HASCIN, bool RELU>
__global__ __launch_bounds__(128)
void k_gemm_wmma(const float* __restrict__ A1, const float* __restrict__ B1,
                 const float* __restrict__ A2, const float* __restrict__ B2,
                 const float* __restrict__ bias, const float* __restrict__ rowScale,
                 const float* __restrict__ Cin, float* __restrict__ D,
                 int M, int K, int Ncol) {
  __shared__ __align__(16) float As1[64][TK + 2];
  __shared__ __align__(16) float As2[64][TK + 2];
  __shared__ __align__(16) float Bs1[TK / 2][64][2];
  __shared__ __align__(16) float Bs2[TK / 2][64][2];

  const int tid  = threadIdx.x;
  const int lane = tid & 31;
  const int wv   = tid >> 5;
  const int tileRow = blockIdx.y * 64;
  const int tileCol = blockIdx.x * 64;
  const bool fullCols = (tileCol + 64 <= Ncol);   // async path needs no zero-fill

  v8f acc[4] = {};

  for (int k0 = 0; k0 < K; k0 += TK) {
    // ---- stage B tiles (16 x 64, K-interleaved): async global->LDS when in bounds
    if (fullCols) {
#pragma unroll
      for (int i = tid; i < TK * 64; i += 128) {
        int r = i >> 6, c = i & 63;
        long gidx = (long)(k0 + r) * Ncol + tileCol + c;
        async_ld_lds_b32(lds_off(&Bs1[r >> 1][c][r & 1]), &B1[gidx]);
        if (DUAL) async_ld_lds_b32(lds_off(&Bs2[r >> 1][c][r & 1]), &B2[gidx]);
      }
    } else {
      for (int i = tid; i < TK * 64; i += 128) {
        int r = i >> 6, c = i & 63;
        int gc = tileCol + c;
        float b = 0.f, b2 = 0.f;
        if (gc < Ncol) {
          b = B1[(long)(k0 + r) * Ncol + gc];
          if (DUAL) b2 = B2[(long)(k0 + r) * Ncol + gc];
        }
        Bs1[r >> 1][c][r & 1] = b;
        if (DUAL) Bs2[r >> 1][c][r & 1] = b2;
      }
    }
    // ---- stage A tiles (64 x 16): scalar path (row guards + optional scale)
    for (int i = tid; i < 64 * TK; i += 128) {
      int r = i >> 4, c = i & 15;
      int gr = tileRow + r;
      float a = 0.f, a2 = 0.f;
      if (gr < M) {
        a = A1[(long)gr * K + k0 + c];
        if (RSCALE) a *= rowScale[gr];
        if (DUAL) a2 = A2[(long)gr * K + k0 + c];
        if (k0 + TK < K) __builtin_prefetch(&A1[(long)gr * K + k0 + TK + c], 0, 1);
      }
      As1[r][c] = a;
      if (DUAL) As2[r][c] = a2;
    }
    wait_async0();
    __syncthreads();

    // ---- WMMA: A frag lane m = lane&15, K-pair = (lane>>4)*2 (ISA 16x4 f32 layout)
    const int mr = wv * 16 + (lane & 15);
    const int kh = (lane >> 4) << 1;
    const int nn = lane & 15;
#pragma unroll
    for (int kk = 0; kk < 4; ++kk) {
      const int kb = kk * 4 + kh;                          // even
      v2f a1 = *(const v2f*)&As1[mr][kb];                  // one ds_load_b64
      v2f a2;
      if (DUAL) a2 = *(const v2f*)&As2[mr][kb];
#pragma unroll
      for (int t = 0; t < 4; ++t) {
        v2f b1 = *(const v2f*)&Bs1[kb >> 1][t * 16 + nn][0];  // one ds_load_b64
        acc[t] = __builtin_amdgcn_wmma_f32_16x16x4_f32(
            false, a1, false, b1, (short)0, acc[t], false, false);
        if (DUAL) {
          v2f b2 = *(const v2f*)&Bs2[kb >> 1][t * 16 + nn][0];
          acc[t] = __builtin_amdgcn_wmma_f32_16x16x4_f32(
              false, a2, false, b2, (short)0, acc[t], false, false);
        }
      }
    }
    __syncthreads();
  }

  // ---- epilogue: C/D layout = 8 VGPRs, rows 0-7 (lanes 0-15) / 8-15 (lanes 16-31)
  const int rbase = tileRow + wv * 16 + ((lane >> 4) << 3);
  const int nc = lane & 15;
#pragma unroll
  for (int t = 0; t < 4; ++t) {
    int colg = tileCol + t * 16 + nc;
    if (colg >= Ncol) continue;
    float bv = bias[colg];
#pragma unroll
    for (int r = 0; r < 8; ++r) {
      int rowg = rbase + r;
      if (rowg < M) {
        float v = acc[t][r] + bv;
        if (HASCIN) v += Cin[(long)rowg * Ncol + colg];
        if (RELU) v = fmaxf(v, 0.f);
        D[(long)rowg * Ncol + colg] = v;
      }
    }
  }
}

// ------------------------------------------------------- tiny final layer (K=96 -> 2)
__global__ void k_classifier(const float* __restrict__ h3, const float* __restrict__ W3,
                             const float* __restrict__ b3, float* __restrict__ out, int N) {
  int n = blockIdx.x * blockDim.x + threadIdx.x;
  if (n >= N) return;
  float a0 = b3[0], a1 = b3[1];
  const float* hr = h3 + (long)n * (HDIM / 4);
#pragma unroll 4
  for (int k = 0; k < HDIM / 4; ++k) {
    float v = hr[k];
    a0 += v * W3[k * NCLS_ + 0];
    a1 += v * W3[k * NCLS_ + 1];
  }
  out[(long)n * NCLS_ + 0] = a0;
  out[(long)n * NCLS_ + 1] = a1;
}

// ---------------------------------------------------------------- launcher
static inline dim3 gemm_grid(int M, int Ncol) {
  return dim3((Ncol + 63) / 64, (M + 63) / 64);
}

extern "C" void kernel_launch(void* const* d_in, const int* in_sizes, int n_in,
                              void* d_out, int out_size, void* d_ws, size_t ws_size,
                              hipStream_t stream) {
  const int N = NNODE, E = NEDGE, H = HDIM;

  const float* x     = (const float*)d_in[0];
  const int*   edge  = (const int*)d_in[1];
  const int*   rowI  = edge;          // edge_index[0]
  const int*   colI  = edge + E;      // edge_index[1]
  const float* W_in  = (const float*)d_in[2];
  const float* b_in  = (const float*)d_in[3];
  const float* bn0_g = (const float*)d_in[4];
  const float* bn0_b = (const float*)d_in[5];
  const float* Wl    = (const float*)d_in[6];
  const float* bl    = (const float*)d_in[7];
  const float* Wr    = (const float*)d_in[8];
  const float* bn_g  = (const float*)d_in[9];
  const float* bn_b  = (const float*)d_in[10];
  const float* Wskip = (const float*)d_in[11];
  const float* bskip = (const float*)d_in[12];
  const float* W1    = (const float*)d_in[13];
  const float* b1    = (const float*)d_in[14];
  const float* W2    = (const float*)d_in[15];
  const float* b2    = (const float*)d_in[16];
  const float* W3    = (const float*)d_in[17];
  const float* b3    = (const float*)d_in[18];

  float* out = (float*)d_out;                 // [N, 2]
  float* emb = out + (size_t)N * NCLS_;       // [N, 384]

  float* ws    = (float*)d_ws;
  float* X0    = ws;
  float* X1    = X0 + (size_t)N * H;
  float* AGG   = X1 + (size_t)N * H;          // also reused for MLP hidden acts
  float* CNT   = AGG + (size_t)N * H;         // [N] -> inv degree
  float* STATS = CNT + N;                     // [2H]

  const dim3 blk1(256);

  // degree -> inverse degree
  k_fill<<<512, blk1, 0, stream>>>(CNT, 0.f, (long)N);
  k_degree<<<(E + 255) / 256, blk1, 0, stream>>>(colI, CNT, E);
  k_invcnt<<<(N + 255) / 256, blk1, 0, stream>>>(CNT, N);

  // input projection: X0 = x @ W_in + b_in
  k_gemm_wmma<false, false, false, false><<<gemm_grid(N, H), 128, 0, stream>>>(
      x, W_in, nullptr, nullptr, b_in, nullptr, nullptr, X0, N, DIN_, H);
  // BN0 + ReLU
  k_fill<<<8, blk1, 0, stream>>>(STATS, 0.f, (long)2 * H);
  k_bn_stats<<<(N + 511) / 512, H, 0, stream>>>(X0, STATS, N);
  k_bn_apply_relu<<<2048, blk1, 0, stream>>>(X0, STATS, bn0_g, bn0_b, N);

  float* X = X0;
  int s = 0;
  for (int i = 0; i < NLAY; ++i) {
    float* T = (X == X0) ? X1 : X0;

    // mean aggregation (scale folded into GEMM A-load)
    k_fill<<<4096, blk1, 0, stream>>>(AGG, 0.f, (long)N * H);
    long scat = (long)E * (H / 4);
    k_scatter<<<(unsigned)((scat + 255) / 256), blk1, 0, stream>>>(X, rowI, colI, AGG, E);

    // fused SAGE layer: T = (AGG * inv_cnt) @ Wl[i] + X @ Wr[i] + bl[i]
    k_gemm_wmma<true, true, false, false><<<gemm_grid(N, H), 128, 0, stream>>>(
        AGG, Wl + (size_t)i * H * H, X, Wr + (size_t)i * H * H,
        bl + (size_t)i * H, CNT, nullptr, T, N, H, H);

    // BN + ReLU
    k_fill<<<8, blk1, 0, stream>>>(STATS, 0.f, (long)2 * H);
    k_bn_stats<<<(N + 511) / 512, H, 0, stream>>>(T, STATS, N);
    k_bn_apply_relu<<<2048, blk1, 0, stream>>>(T, STATS, bn_g + (size_t)i * H,
                                               bn_b + (size_t)i * H, N);

    if ((i + 1) % 2 == 0) {
      // residual skip: Dst = T + identity @ Wskip[s] + bskip[s]
      float* Dst = (i == NLAY - 1) ? emb : T;   // final layer writes embeddings to d_out
      k_gemm_wmma<false, false, true, false><<<gemm_grid(N, H), 128, 0, stream>>>(
          X, Wskip + (size_t)s * H * H, nullptr, nullptr,
          bskip + (size_t)s * H, nullptr, T, Dst, N, H, H);
      ++s;
      X = Dst;
    } else {
      X = T;
    }
  }

  // MLP head (reuse AGG region for hidden activations)
  float* H2 = AGG;                               // [N, 192]
  float* H3 = AGG + (size_t)N * (H / 2);         // [N, 96]
  k_gemm_wmma<false, false, false, true><<<gemm_grid(N, H / 2), 128, 0, stream>>>(
      X, W1, nullptr, nullptr, b1, nullptr, nullptr, H2, N, H, H / 2);
  k_gemm_wmma<false, false, false, true><<<gemm_grid(N, H / 4), 128, 0, stream>>>(
      H2, W2, nullptr, nullptr, b2, nullptr, nullptr, H3, N, H / 2, H / 4);
  k_classifier<<<(N + 255) / 256, blk1, 0, stream>>>(H3, W3, b3, out, N);
}